// GNNLayer_24335284699303
// MI455X (gfx1250) — compile-verified
//
#include <hip/hip_runtime.h>

#define DIM 128
#define HID 512
#define BN_EPS 1e-5f

// GEMM tiling: block = 256 threads = 8 waves; 128x64 output tile; K chunked by 32,
// double-buffered LDS staging via the CDNA5 async global->LDS path (ASYNCcnt).
#define BM 128
#define BN 64
#define BK 32
#define AS_STRIDE 36   // floats; 144B rows: 16B-aligned chunks, bank-spread b64 reads
#define BS_STRIDE 68   // floats; 272B rows: 16B-aligned chunks
#define A_ELEMS (BM * AS_STRIDE)     // 4608 floats / 18,432 B per buffer
#define B_ELEMS (BK * BS_STRIDE)     // 2176 floats /  8,704 B per buffer
#define STAGE_OPS 6                  // async b128 ops per thread per chunk (4 A + 2 B)

typedef float v2f __attribute__((ext_vector_type(2)));
typedef float v8f __attribute__((ext_vector_type(8)));

// ---------------------------------------------------------------------------
// CDNA5 async global->LDS copy (ASYNCcnt path), 16B per lane.
// ---------------------------------------------------------------------------
__device__ __forceinline__ void async_g2lds_b128(const float* gptr, unsigned int lds_byte_off) {
    unsigned long long ga = (unsigned long long)(uintptr_t)gptr;
    asm volatile("global_load_async_to_lds_b128 %0, %1, off"
                 :: "v"(lds_byte_off), "v"(ga)
                 : "memory");
}
// Async loads complete in order: waiting ASYNCcnt<=STAGE_OPS after issuing the
// next chunk's STAGE_OPS guarantees the current chunk's fills are done.
__device__ __forceinline__ void wait_async_le_stage() {
    asm volatile("s_wait_asynccnt 0x6" ::: "memory");
}
__device__ __forceinline__ void wait_async_le0() {
    asm volatile("s_wait_asynccnt 0x0" ::: "memory");
}

// ---------------------------------------------------------------------------
// WMMA fp32 GEMM: C = act(A[M,K] @ B[K,N] + bias) (+ residual)
// Double-buffered async global->LDS staging overlapped with V_WMMA_F32_16X16X4_F32.
// Requires: K % 32 == 0, Ncols % 64 == 0, M % 16 == 0.
// ---------------------------------------------------------------------------
__global__ void __launch_bounds__(256) wmma_gemm_f32(const float* __restrict__ A,
                                                     const float* __restrict__ B,
                                                     const float* __restrict__ bias,
                                                     const float* __restrict__ residual,
                                                     float* __restrict__ C,
                                                     int M, int Ncols, int K, int do_relu)
{
    __shared__ float As[2][A_ELEMS];   // 2 x 18,432 B
    __shared__ float Bs[2][B_ELEMS];   // 2 x  8,704 B   (total ~54 KB)

    const int tid  = threadIdx.x;
    const int wv   = tid >> 5;
    const int lane = tid & 31;

    const int nbn  = Ncols / BN;
    const int bm   = blockIdx.x / nbn;
    const int bn   = blockIdx.x - bm * nbn;
    const int row0 = bm * BM + wv * 16;            // this wave's 16-row slice
    const bool valid = (row0 + 16) <= M;           // store guard (barriers still run)
    const int col0 = bn * BN;

    const int mr = lane & 15;                      // A row in tile / B,C column idx
    const int kb = (lane >> 4) << 1;               // 0 or 2

    const unsigned int asBase = (unsigned int)(uintptr_t)&As[0][0];
    const unsigned int bsBase = (unsigned int)(uintptr_t)&Bs[0][0];

    // Stage one K-chunk of A (128 x 32 floats, 1024 x 16B) and B (32 x 64, 512 x 16B).
    auto stage = [&](int buf, int kc) {
        #pragma unroll
        for (int i = 0; i < 4; ++i) {              // A: 4 chunks / thread
            int cid = tid + i * 256;
            int r   = cid >> 3;
            int c4  = (cid & 7) << 2;
            int rg  = bm * BM + r;  if (rg > M - 1) rg = M - 1;   // clamp, no OOB
            async_g2lds_b128(A + (size_t)rg * K + kc + c4,
                             asBase + (unsigned int)(buf * A_ELEMS + r * AS_STRIDE + c4) * 4u);
        }
        #pragma unroll
        for (int i = 0; i < 2; ++i) {              // B: 2 chunks / thread
            int cid = tid + i * 256;
            int r   = cid >> 4;
            int c4  = (cid & 15) << 2;
            async_g2lds_b128(B + (size_t)(kc + r) * Ncols + col0 + c4,
                             bsBase + (unsigned int)(buf * B_ELEMS + r * BS_STRIDE + c4) * 4u);
        }
    };

    v8f acc[4] = {};
    const int nchunk = K / BK;

    stage(0, 0);
    for (int ci = 0; ci < nchunk; ++ci) {
        const int cur = ci & 1;
        if (ci + 1 < nchunk) {
            stage(cur ^ 1, (ci + 1) * BK);         // fill next buffer while we compute
            wait_async_le_stage();                 // current buffer's 6 ops complete
        } else {
            wait_async_le0();
        }
        __syncthreads();

        const float* Arow = &As[cur][0] + (wv * 16 + mr) * AS_STRIDE + kb;
        const float* Brow = &Bs[cur][0] + kb * BS_STRIDE + mr;
        #pragma unroll
        for (int k = 0; k < BK; k += 4) {
            float2 av = *(const float2*)(Arow + k);        // ds_load_b64
            v2f a; a.x = av.x; a.y = av.y;
            const float* bp = Brow + k * BS_STRIDE;
            #pragma unroll
            for (int j = 0; j < 4; ++j) {
                v2f b; b.x = bp[j * 16]; b.y = bp[BS_STRIDE + j * 16];
                acc[j] = __builtin_amdgcn_wmma_f32_16x16x4_f32(
                    false, a, false, b, (short)0, acc[j], false, false);
            }
        }
        __syncthreads();                           // reads done before buffer is refilled
    }

    if (!valid) return;
    // C tile: VGPR v holds M = v + 8*(lane>>4), N = lane&15
    const int mofs = (lane >> 4) << 3;
    #pragma unroll
    for (int j = 0; j < 4; ++j) {
        const int col = col0 + (j << 4) + mr;
        const float bv = bias ? bias[col] : 0.0f;
        #pragma unroll
        for (int v = 0; v < 8; ++v) {
            const int r = row0 + mofs + v;
            float val = acc[j][v] + bv;
            if (do_relu) val = fmaxf(val, 0.0f);
            const size_t idx = (size_t)r * Ncols + col;
            if (residual) val += residual[idx];
            C[idx] = val;
        }
    }
}

// ---------------------------------------------------------------------------
// Degree / normalization kernels
// ---------------------------------------------------------------------------
__global__ void deg_init(float* __restrict__ deg, int n) {
    int i = blockIdx.x * blockDim.x + threadIdx.x;
    if (i < n) deg[i] = 1.0f;                // self-loop contribution
}

__global__ void deg_accum(const int* __restrict__ dst, float* __restrict__ deg, int e) {
    int i = blockIdx.x * blockDim.x + threadIdx.x;
    if (i < e) atomicAdd(&deg[dst[i]], 1.0f);
}

__global__ void deg_rsqrt(float* __restrict__ deg, int n) {
    int i = blockIdx.x * blockDim.x + threadIdx.x;
    if (i < n) deg[i] = rsqrtf(deg[i]);      // deg >= 1 always
}

// h[i,:] = xw[i,:] * dinv[i]^2 + b   (self-loop message + GCN bias, initializes h)
__global__ void self_loop(const float* __restrict__ xw, const float* __restrict__ dinv,
                          const float* __restrict__ bias, float* __restrict__ h, int n) {
    int t = blockIdx.x * blockDim.x + threadIdx.x;
    int node = t >> 5, lane = t & 31;
    if (node >= n) return;
    float di = dinv[node];
    float nrm = di * di;
    float4 v  = ((const float4*)(xw + (size_t)node * DIM))[lane];
    float4 bv = ((const float4*)bias)[lane];
    float4 o  = { v.x * nrm + bv.x, v.y * nrm + bv.y,
                  v.z * nrm + bv.z, v.w * nrm + bv.w };
    ((float4*)(h + (size_t)node * DIM))[lane] = o;
}

// One wave per edge, grid-stride. xw/h (51.2 MB each) are L2-resident vs the
// 192 MB L2, so gathers + atomic scatter-adds stay on-chip; prefetch the next
// iteration's source row (global_prefetch_b8) to hide gather latency.
__global__ void scatter_edges(const int* __restrict__ src, const int* __restrict__ dst,
                              const float* __restrict__ dinv, const float* __restrict__ xw,
                              float* __restrict__ h, int e) {
    int wave   = (blockIdx.x * blockDim.x + threadIdx.x) >> 5;
    int lane   = threadIdx.x & 31;
    int nwaves = (gridDim.x * blockDim.x) >> 5;
    for (int edge = wave; edge < e; edge += nwaves) {
        int nxt = edge + nwaves;
        if (nxt < e) {
            int sp = src[nxt];
            __builtin_prefetch(xw + (size_t)sp * DIM + lane * 4, 0, 0);
        }
        int s = src[edge], d = dst[edge];
        float nrm = dinv[s] * dinv[d];
        float4 v = ((const float4*)(xw + (size_t)s * DIM))[lane];
        float* hd = h + (size_t)d * DIM + lane * 4;
        atomicAdd(hd + 0, v.x * nrm);
        atomicAdd(hd + 1, v.y * nrm);
        atomicAdd(hd + 2, v.z * nrm);
        atomicAdd(hd + 3, v.w * nrm);
    }
}

// ---------------------------------------------------------------------------
// BatchNorm (training mode, biased variance) — fused residual add + stats pass
// ---------------------------------------------------------------------------
__global__ void bn_stats(const float* __restrict__ a, const float* __restrict__ bb,
                         float* __restrict__ z, float* __restrict__ stats, int n) {
    const int c = threadIdx.x;               // column 0..127
    float s = 0.f, sq = 0.f;
    for (int r = blockIdx.x * blockDim.y + threadIdx.y; r < n; r += gridDim.x * blockDim.y) {
        size_t i = (size_t)r * DIM + c;
        float v = a[i];
        if (bb) v += bb[i];
        z[i] = v;
        s += v; sq += v * v;
    }
    __shared__ float ls[4][DIM];
    __shared__ float lq[4][DIM];
    ls[threadIdx.y][c] = s;
    lq[threadIdx.y][c] = sq;
    __syncthreads();
    if (threadIdx.y == 0) {
        for (int y = 1; y < 4; ++y) { s += ls[y][c]; sq += lq[y][c]; }
        atomicAdd(&stats[c], s);
        atomicAdd(&stats[DIM + c], sq);
    }
}

__global__ void bn_finalize(const float* __restrict__ stats, const float* __restrict__ gamma,
                            const float* __restrict__ beta, float* __restrict__ scsh, float inv_n) {
    int c = threadIdx.x;
    float mean = stats[c] * inv_n;
    float var  = stats[DIM + c] * inv_n - mean * mean;
    float sc   = gamma[c] * rsqrtf(var + BN_EPS);
    scsh[c]        = sc;
    scsh[DIM + c]  = beta[c] - mean * sc;
}

__global__ void bn_apply(const float* __restrict__ z, const float* __restrict__ scsh,
                         float* __restrict__ out, int n) {
    int t = blockIdx.x * blockDim.x + threadIdx.x;
    int node = t >> 5, lane = t & 31;
    if (node >= n) return;
    float4 v  = ((const float4*)(z + (size_t)node * DIM))[lane];
    float4 sc = ((const float4*)(scsh))[lane];
    float4 sh = ((const float4*)(scsh + DIM))[lane];
    float4 o  = { v.x * sc.x + sh.x, v.y * sc.y + sh.y,
                  v.z * sc.z + sh.z, v.w * sc.w + sh.w };
    ((float4*)(out + (size_t)node * DIM))[lane] = o;
}

// ---------------------------------------------------------------------------
static inline void launch_gemm(const float* A, const float* B, const float* bias,
                               const float* res, float* C, int M, int Nc, int K,
                               int relu, hipStream_t s) {
    int mb = (M + BM - 1) / BM;
    int nb = Nc / BN;
    hipLaunchKernelGGL(wmma_gemm_f32, dim3(mb * nb), dim3(256), 0, s,
                       A, B, bias, res, C, M, Nc, K, relu);
}

extern "C" void kernel_launch(void* const* d_in, const int* in_sizes, int n_in,
                              void* d_out, int out_size, void* d_ws, size_t ws_size,
                              hipStream_t stream) {
    (void)n_in; (void)out_size; (void)ws_size;
    const float* x     = (const float*)d_in[0];
    const int*   eidx  = (const int*)  d_in[1];   // [2,E]: row0=src, row1=dst
    const float* W     = (const float*)d_in[2];
    const float* b     = (const float*)d_in[3];
    const float* gamma = (const float*)d_in[4];
    const float* beta  = (const float*)d_in[5];
    const float* w1    = (const float*)d_in[6];
    const float* b1    = (const float*)d_in[7];
    const float* w2    = (const float*)d_in[8];
    const float* b2    = (const float*)d_in[9];
    float* out = (float*)d_out;

    const int n = in_sizes[0] / DIM;              // 100000 (multiple of 16)
    const int e = in_sizes[1] / 2;                // 3200000
    const int* src = eidx;
    const int* dst = eidx + e;

    // Workspace carve-out
    char* ws = (char*)d_ws;
    float* bufA  = (float*)ws;                          ws += (size_t)n * DIM * sizeof(float); // xw -> y1
    float* bufB  = (float*)ws;                          ws += (size_t)n * DIM * sizeof(float); // h -> z
    float* bufC  = (float*)ws;                          ws += (size_t)n * HID * sizeof(float); // hidden
    float* deg   = (float*)ws;                          ws += (size_t)n * sizeof(float);       // deg -> dinv
    float* stats = (float*)ws;                          ws += 2 * DIM * sizeof(float);
    float* scsh  = (float*)ws;                          ws += 2 * DIM * sizeof(float);

    const int blk = 256;
    const dim3 bnBlk(DIM, 4);
    const int bnGrid = 1024;

    // --- GCN conv ---
    hipLaunchKernelGGL(deg_init,  dim3((n + blk - 1) / blk), dim3(blk), 0, stream, deg, n);
    hipLaunchKernelGGL(deg_accum, dim3((e + blk - 1) / blk), dim3(blk), 0, stream, dst, deg, e);
    hipLaunchKernelGGL(deg_rsqrt, dim3((n + blk - 1) / blk), dim3(blk), 0, stream, deg, n);

    launch_gemm(x, W, nullptr, nullptr, bufA, n, DIM, DIM, 0, stream);         // xw = x@W

    {
        long long th = (long long)n * 32;
        hipLaunchKernelGGL(self_loop, dim3((int)((th + blk - 1) / blk)), dim3(blk), 0, stream,
                           bufA, deg, b, bufB, n);                              // h = xw*dinv^2 + b
    }
    hipLaunchKernelGGL(scatter_edges, dim3(4096), dim3(blk), 0, stream,
                       src, dst, deg, bufA, bufB, e);                           // h += norm * xw[src]

    // --- BN1: y1 = BN(x + h) ---
    hipMemsetAsync(stats, 0, 2 * DIM * sizeof(float), stream);
    hipLaunchKernelGGL(bn_stats, dim3(bnGrid), bnBlk, 0, stream, x, bufB, bufB, stats, n); // z = x + h
    hipLaunchKernelGGL(bn_finalize, dim3(1), dim3(DIM), 0, stream, stats, gamma, beta, scsh, 1.0f / n);
    {
        long long th = (long long)n * 32;
        hipLaunchKernelGGL(bn_apply, dim3((int)((th + blk - 1) / blk)), dim3(blk), 0, stream,
                           bufB, scsh, bufA, n);                                // y1 (overwrites xw)
    }

    // --- FF: z2 = y1 + relu(y1@w1 + b1)@w2 + b2 ---
    launch_gemm(bufA, w1, b1, nullptr, bufC, n, HID, DIM, 1, stream);           // hidden
    launch_gemm(bufC, w2, b2, bufA, bufB, n, DIM, HID, 0, stream);              // z2 (+residual y1)

    // --- BN2 -> out ---
    hipMemsetAsync(stats, 0, 2 * DIM * sizeof(float), stream);
    hipLaunchKernelGGL(bn_stats, dim3(bnGrid), bnBlk, 0, stream, bufB, (const float*)nullptr, bufB, stats, n);
    hipLaunchKernelGGL(bn_finalize, dim3(1), dim3(DIM), 0, stream, stats, gamma, beta, scsh, 1.0f / n);
    {
        long long th = (long long)n * 32;
        hipLaunchKernelGGL(bn_apply, dim3((int)((th + blk - 1) / blk)), dim3(blk), 0, stream,
                           bufB, scsh, out, n);
    }
}